// QuantumLayer_78692390797439
// MI455X (gfx1250) — compile-verified
//
#include <hip/hip_runtime.h>
#include <math.h>

typedef __attribute__((ext_vector_type(2))) float v2f;
typedef __attribute__((ext_vector_type(8))) float v8f;

#define NQ  4
#define DIM 16

// ---------------------------------------------------------------------------
// Kernel 1: build the fixed 16x16 complex unitary U from the 24 weights.
// One block, 256 threads, U kept in LDS as separate re/im planes.
// Gate on qubit q acts on bit position (3-q); RX/RY/RZ fused in registers
// per qubit (same pair partition -> no barrier between the three rotations).
// ---------------------------------------------------------------------------
__global__ __launch_bounds__(256) void build_unitary(const float* __restrict__ wts,
                                                     float* __restrict__ ws) {
    __shared__ float Ur[DIM][DIM];
    __shared__ float Ui[DIM][DIM];
    const int t = threadIdx.x;
    const int i = t >> 4, j = t & 15;
    Ur[i][j] = (i == j) ? 1.0f : 0.0f;
    Ui[i][j] = 0.0f;
    __syncthreads();

    for (int l = 0; l < 2; ++l) {
        for (int q = 0; q < NQ; ++q) {
            const int bitpos = 3 - q;
            if (t < 128) {
                const int col = t & 15;
                const int p   = t >> 4;                          // 0..7
                const int i0  = ((p >> bitpos) << (bitpos + 1)) | (p & ((1 << bitpos) - 1));
                const int i1  = i0 | (1 << bitpos);
                float u0r = Ur[i0][col], u0i = Ui[i0][col];
                float u1r = Ur[i1][col], u1i = Ui[i1][col];
                const float tx = wts[(l * NQ + q) * 3 + 0];
                const float ty = wts[(l * NQ + q) * 3 + 1];
                const float tz = wts[(l * NQ + q) * 3 + 2];
                // RX(tx): [[c, -i s], [-i s, c]]
                float c = cosf(0.5f * tx), s = sinf(0.5f * tx);
                float n0r = c * u0r + s * u1i, n0i = c * u0i - s * u1r;
                float n1r = c * u1r + s * u0i, n1i = c * u1i - s * u0r;
                u0r = n0r; u0i = n0i; u1r = n1r; u1i = n1i;
                // RY(ty): [[c, -s], [s, c]] (real)
                c = cosf(0.5f * ty); s = sinf(0.5f * ty);
                n0r = c * u0r - s * u1r; n0i = c * u0i - s * u1i;
                n1r = s * u0r + c * u1r; n1i = s * u0i + c * u1i;
                u0r = n0r; u0i = n0i; u1r = n1r; u1i = n1i;
                // RZ(tz): diag(e^{-i tz/2}, e^{+i tz/2})
                c = cosf(0.5f * tz); s = sinf(0.5f * tz);
                n0r = c * u0r + s * u0i; n0i = c * u0i - s * u0r;
                n1r = c * u1r - s * u1i; n1i = c * u1i + s * u1r;
                Ur[i0][col] = n0r; Ui[i0][col] = n0i;
                Ur[i1][col] = n1r; Ui[i1][col] = n1i;
            }
            __syncthreads();
        }
        // CNOT ring: new_state[idx] = old_state[perm[idx]]
        for (int q = 0; q < NQ; ++q) {
            const int tq    = (q + 1) & 3;
            const int bitC  = 3 - q;
            const int maskT = 1 << (3 - tq);
            const int src   = ((i >> bitC) & 1) ? (i ^ maskT) : i;
            const float tr = Ur[src][j], ti = Ui[src][j];
            __syncthreads();
            Ur[i][j] = tr; Ui[i][j] = ti;
            __syncthreads();
        }
    }
    // Row-major: ws[0..255] = Re(U), ws[256..511] = Im(U)
    ws[i * DIM + j]       = Ur[i][j];
    ws[256 + i * DIM + j] = Ui[i][j];
}

// ---------------------------------------------------------------------------
// Kernel 2: per 16-element batch tile, compute V = S0 * U^T (re & im) with
// V_WMMA_F32_16X16X4_F32 (K=16 -> 4 chained WMMAs per accumulator), then
// probs = |V|^2 and out = probs @ Z^T via a 4-stage Walsh-Hadamard shfl_xor
// butterfly (Z rows are exactly WHT rows at indices 1,2,4,8).
// ---------------------------------------------------------------------------
__global__ __launch_bounds__(256) void qlayer_wmma(const float* __restrict__ x,
                                                   const float* __restrict__ U,
                                                   float* __restrict__ out, int B) {
    const int lane  = threadIdx.x & 31;
    const int hi    = lane >> 4;        // half-wave: 0 -> lanes 0-15, 1 -> 16-31
    const int n     = lane & 15;        // N index (output state) / M index (batch row)
    const int o     = hi * 2;           // K sub-offset within a 4-wide chunk

    const float* Ure = U;
    const float* Uim = U + 256;

    // B fragments: B[k][n] = U[n][k]; chunk t covers k = 4t..4t+3.
    // Lane layout mirrors A: lanes 0-15 hold k0,k0+1; lanes 16-31 hold k0+2,k0+3.
    v2f bre[4], bim[4];
#pragma unroll
    for (int t = 0; t < 4; ++t) {
        bre[t] = *reinterpret_cast<const v2f*>(Ure + n * 16 + 4 * t + o);
        bim[t] = *reinterpret_cast<const v2f*>(Uim + n * 16 + 4 * t + o);
    }

    const int wavesPerBlock = blockDim.x >> 5;
    const int gwave  = blockIdx.x * wavesPerBlock + (threadIdx.x >> 5);
    const int nwaves = gridDim.x * wavesPerBlock;
    const int ntiles = B >> 4;          // BATCH is a multiple of 16

    for (int tile = gwave; tile < ntiles; tile += nwaves) {
        const int base = tile << 4;
        // A fragment: row m = n (both half-waves load the same rows).
        const float4 xv = *reinterpret_cast<const float4*>(x + (size_t)(base + n) * 4);
        float c0, s0, c1, s1, c2, s2, c3, s3;
        __sincosf(0.5f * xv.x, &s0, &c0);
        __sincosf(0.5f * xv.y, &s1, &c1);
        __sincosf(0.5f * xv.z, &s2, &c2);
        __sincosf(0.5f * xv.w, &s3, &c3);
        // state index k bits: bit3<-qubit0, bit2<-qubit1, bit1<-qubit2, bit0<-qubit3
        // For chunk t: bit3 = t>>1, bit2 = t&1 (compile-time); bit1 = hi; bit0 = j.
        const float f2 = hi ? s2 : c2;
        const float fx = f2 * c3;                 // j = 0
        const float fy = f2 * s3;                 // j = 1
        const float g00 = c0 * c1, g01 = c0 * s1, g10 = s0 * c1, g11 = s0 * s1;
        v2f a[4];
        a[0].x = g00 * fx; a[0].y = g00 * fy;
        a[1].x = g01 * fx; a[1].y = g01 * fy;
        a[2].x = g10 * fx; a[2].y = g10 * fy;
        a[3].x = g11 * fx; a[3].y = g11 * fy;

        v8f cre = {}; v8f cim = {};
#pragma unroll
        for (int t = 0; t < 4; ++t) {
            cre = __builtin_amdgcn_wmma_f32_16x16x4_f32(false, a[t], false, bre[t],
                                                        (short)0, cre, false, false);
            cim = __builtin_amdgcn_wmma_f32_16x16x4_f32(false, a[t], false, bim[t],
                                                        (short)0, cim, false, false);
        }

        // probs, then WHT butterfly over the 16-lane halves (masks 1,2,4,8 stay
        // within each half). Lane q = lane&15 ends with F[q] = sum_i (-1)^{popc(q&i)} p[i].
        float p[8];
#pragma unroll
        for (int r = 0; r < 8; ++r) p[r] = cre[r] * cre[r] + cim[r] * cim[r];
#pragma unroll
        for (int s = 1; s <= 8; s <<= 1) {
#pragma unroll
            for (int r = 0; r < 8; ++r) {
                const float tv = __shfl_xor(p[r], s, 32);
                p[r] = (lane & s) ? (tv - p[r]) : (p[r] + tv);
            }
        }
        // out[m][w] = F[1 << (3-w)];  m = r + 8*hi
        const int q = lane & 15;
        if (q == 1 || q == 2 || q == 4 || q == 8) {
            const int w = 3 - (31 - __clz(q));
#pragma unroll
            for (int r = 0; r < 8; ++r) {
                const int gb = base + r + 8 * hi;
                out[(size_t)gb * 4 + w] = p[r];
            }
        }
    }
}

extern "C" void kernel_launch(void* const* d_in, const int* in_sizes, int n_in,
                              void* d_out, int out_size, void* d_ws, size_t ws_size,
                              hipStream_t stream) {
    const float* x   = (const float*)d_in[0];   // (BATCH, 4) f32
    const float* wts = (const float*)d_in[1];   // (2, 4, 3)  f32
    float* out = (float*)d_out;                 // (BATCH, 4) f32
    float* ws  = (float*)d_ws;                  // needs 512 floats (2 KB)
    const int B = in_sizes[0] / 4;

    build_unitary<<<1, 256, 0, stream>>>(wts, ws);
    qlayer_wmma<<<2048, 256, 0, stream>>>(x, ws, out, B);
}